// RGCN_35854386987109
// MI455X (gfx1250) — compile-verified
//
#include <hip/hip_runtime.h>
#include <hip/hip_bf16.h>

// ---------------------------------------------------------------------------
// RGCN (2 conv layers + linear head) for MI455X / gfx1250.
//
// Math identity: mean_e(W[r] x[src_e]) = W[r] mean_e(x[src_e]) (transform is
// linear, aggregation is a mean) =>
//   h[n] = b + x[n] @ root + sum_r ( (agg[n,r]/max(cnt[n,r],1)) @ W[r] )
// agg is a per-(dst,rel) scatter-add of raw source features.
//
// Dense transforms run on V_WMMA_F32_16X16X4_F32 (exact f32, matches the f32
// reference). GEMM data movement:
//   - weights pre-transposed once (19 x 64KB) so B loads are contiguous b64,
//     then held in registers (32 x v2f / wave) across 5 row tiles per block
//   - A tile staged in LDS (stride 132: conflict-free ds_load_b64, aligned
//     ds_store_b128), with the per-row mean scale folded into staging
// Inner loop: 1 ds_load_b64 + 1 v_wmma per 2048 FLOP.
// Scatter/count use f32 atomics; x/agg (25.6 MB each) are L2-resident (192MB).
// ---------------------------------------------------------------------------

#define NN 50000
#define NE 1000000
#define CH 128
#define NREL 8
#define ROWT 5                      // row tiles per block; 3125 = 625 * 5
#define NBLK 625
#define LDSTRIDE 132                // floats per staged A row (bank-conflict pad)

typedef float v2f __attribute__((ext_vector_type(2)));
typedef float v8f __attribute__((ext_vector_type(8)));

__global__ void rgcn_zero_kernel(float* __restrict__ p, int n) {
    for (int i = blockIdx.x * blockDim.x + threadIdx.x; i < n;
         i += gridDim.x * blockDim.x)
        p[i] = 0.0f;
}

__global__ void rgcn_count_kernel(const int* __restrict__ dst,
                                  const int* __restrict__ et,
                                  float* __restrict__ cnt) {
    int e = blockIdx.x * blockDim.x + threadIdx.x;
    if (e < NE) atomicAdd(&cnt[dst[e] * NREL + et[e]], 1.0f);
}

// out[m][n][k] = in[m][k][n] for nmats 128x128 matrices (one-time, ~1.2MB total)
__global__ void rgcn_transpose_kernel(const float* __restrict__ in,
                                      float* __restrict__ out, int nmats) {
    int total = nmats << 14;
    for (int i = blockIdx.x * blockDim.x + threadIdx.x; i < total;
         i += gridDim.x * blockDim.x) {
        int m = i >> 14, e = i & 16383;
        int k = e >> 7, n = e & 127;
        out[(m << 14) + n * CH + k] = in[(m << 14) + k * CH + n];
    }
}

// One wave per edge; waves whose edge is not of relation `rel` exit early
// (et[e] is wave-uniform -> scalar branch, cheap skip).
__global__ void rgcn_scatter_kernel(const float* __restrict__ x,
                                    const int* __restrict__ src,
                                    const int* __restrict__ dst,
                                    const int* __restrict__ et,
                                    int rel,
                                    float* __restrict__ agg) {
    int gtid = blockIdx.x * blockDim.x + threadIdx.x;
    int e    = gtid >> 5;
    int lane = gtid & 31;
    if (e >= NE) return;
    if (et[e] != rel) return;
    const float* xs = x + (size_t)src[e] * CH;
    float*       ad = agg + (size_t)dst[e] * CH;
#pragma unroll
    for (int c0 = 0; c0 < CH; c0 += 32)
        atomicAdd(ad + c0 + lane, xs[c0 + lane]);
}

__global__ void rgcn_relu_kernel(float* __restrict__ h, int n) {
    int i = blockIdx.x * blockDim.x + threadIdx.x;
    if (i < n) h[i] = fmaxf(h[i], 0.0f);
}

// ---------------------------------------------------------------------------
// WMMA f32 GEMM: out (+)= (A * rowscale) @ Bt^T [+ bias]
//   A   : [NN x 128] row-major
//   Bt  : [128 x 128] TRANSPOSED weight ([n][k]) -> contiguous b64 per lane
//   bias: non-null => C = broadcast bias (overwrite); null => C = out (accum)
//   cnt : non-null => A row m scaled by 1/max(cnt[m*8+rel],1) (mean agg)
// Block (32,8): wave y owns cols [16y,16y+16); block covers ROWT row tiles.
// All control flow is wave-uniform -> EXEC all-ones at every WMMA (ISA req).
//
// VGPR layouts per CDNA5 ISA (16x16x4 f32):
//   A: lane l holds row (l&15); v2f = K {2*(l>>4), 2*(l>>4)+1}
//   B: lane l holds col (l&15); v2f = same K pair
//   C/D: c[v] = element (row 8*(l>>4)+v, col l&15)
// ---------------------------------------------------------------------------
__global__ __launch_bounds__(256) void rgcn_gemm_wmma(
    const float* __restrict__ A,
    const float* __restrict__ Bt,
    const float* __restrict__ bias,
    const float* __restrict__ cnt,
    int rel,
    float* __restrict__ out) {
    __shared__ float a_lds[16 * LDSTRIDE];

    const int lane    = threadIdx.x;          // 0..31
    const int nloc    = lane & 15;
    const int kgrp    = lane >> 4;            // 0/1
    const int colbase = threadIdx.y * 16;
    const int tid     = threadIdx.y * 32 + lane;

    // Wave's B column tile -> registers: breg[k4] = Bt[col][4*k4 + 2*kgrp +{0,1}]
    v2f breg[32];
    {
        const float* bp = Bt + (size_t)(colbase + nloc) * CH + 2 * kgrp;
#pragma unroll
        for (int i = 0; i < 32; ++i)
            breg[i] = *(const v2f*)(bp + i * 4);
    }

    float bv = 0.0f;
    if (bias) bv = bias[colbase + nloc];

    for (int t = 0; t < ROWT; ++t) {
        const int rowbase = (blockIdx.x * ROWT + t) * 16;

        __syncthreads();   // previous tile's LDS readers are done
        // Cooperatively stage the 16x128 A tile, folding in the mean scale.
        // 512 float4 chunks, 256 threads -> 2 chunks each; b128 loads/stores.
#pragma unroll
        for (int j = tid; j < 512; j += 256) {
            int row = j >> 5;
            int col = (j & 31) * 4;
            float s = 1.0f;
            if (cnt) s = 1.0f / fmaxf(cnt[(rowbase + row) * NREL + rel], 1.0f);
            const float4 g = *(const float4*)(A + (size_t)(rowbase + row) * CH + col);
            float* la = &a_lds[row * LDSTRIDE + col];
            la[0] = g.x * s; la[1] = g.y * s; la[2] = g.z * s; la[3] = g.w * s;
        }
        __syncthreads();

        v8f c;
        if (bias) {
#pragma unroll
            for (int v = 0; v < 8; ++v) c[v] = bv;
        } else {
#pragma unroll
            for (int v = 0; v < 8; ++v)
                c[v] = out[(size_t)(rowbase + kgrp * 8 + v) * CH + colbase + nloc];
        }

        const float* la = &a_lds[nloc * LDSTRIDE + 2 * kgrp];
#pragma unroll
        for (int k = 0; k < 32; ++k) {
            v2f a = *(const v2f*)(la + k * 4);     // ds_load_b64, conflict-free
            // (neg_a, A, neg_b, B, c_mod, C, reuse_a, reuse_b)
            c = __builtin_amdgcn_wmma_f32_16x16x4_f32(false, a, false, breg[k],
                                                      (short)0, c, false, false);
        }

#pragma unroll
        for (int v = 0; v < 8; ++v)
            out[(size_t)(rowbase + kgrp * 8 + v) * CH + colbase + nloc] = c[v];
    }
}

// ---------------------------------------------------------------------------
static void run_layer(const float* x_in, const float* Wt, const float* rootT,
                      const float* b, const int* src, const int* dst,
                      const int* et, const float* cnt, float* agg,
                      float* h_out, hipStream_t stream) {
    dim3 gblk(32, 8);
    // h = b + x @ root
    rgcn_gemm_wmma<<<NBLK, gblk, 0, stream>>>(x_in, rootT, b, nullptr, 0, h_out);
    // h += sum_r mean-agg_r @ W[r]
    for (int r = 0; r < NREL; ++r) {
        rgcn_zero_kernel<<<2048, 256, 0, stream>>>(agg, NN * CH);
        rgcn_scatter_kernel<<<(NE * 32) / 256, 256, 0, stream>>>(
            x_in, src, dst, et, r, agg);
        rgcn_gemm_wmma<<<NBLK, gblk, 0, stream>>>(
            agg, Wt + (size_t)r * CH * CH, nullptr, cnt, r, h_out);
    }
}

extern "C" void kernel_launch(void* const* d_in, const int* in_sizes, int n_in,
                              void* d_out, int out_size, void* d_ws, size_t ws_size,
                              hipStream_t stream) {
    const float* x     = (const float*)d_in[0];
    const int*   ei    = (const int*)d_in[1];
    const int*   et    = (const int*)d_in[2];
    const float* W1    = (const float*)d_in[3];
    const float* root1 = (const float*)d_in[4];
    const float* b1    = (const float*)d_in[5];
    const float* W2    = (const float*)d_in[6];
    const float* root2 = (const float*)d_in[7];
    const float* b2    = (const float*)d_in[8];
    const float* linW  = (const float*)d_in[9];
    const float* linb  = (const float*)d_in[10];
    (void)in_sizes; (void)n_in; (void)out_size; (void)ws_size;

    const int* src = ei;        // ei[0, :]
    const int* dst = ei + NE;   // ei[1, :]

    // ws layout (floats):
    // cnt[NN*8] | agg[NN*128] | h1[NN*128] | h2[NN*128] |
    // W1t[8*16384] | root1t[16384] | W2t[8*16384] | root2t[16384] | linWt[16384]
    float* cnt    = (float*)d_ws;
    float* agg    = cnt    + (size_t)NN * NREL;
    float* h1     = agg    + (size_t)NN * CH;
    float* h2     = h1     + (size_t)NN * CH;
    float* W1t    = h2     + (size_t)NN * CH;
    float* root1t = W1t    + (size_t)NREL * CH * CH;
    float* W2t    = root1t + (size_t)CH * CH;
    float* root2t = W2t    + (size_t)NREL * CH * CH;
    float* linWt  = root2t + (size_t)CH * CH;

    // One-time weight transposes (k-major -> n-major)
    rgcn_transpose_kernel<<<512, 256, 0, stream>>>(W1, W1t, NREL);
    rgcn_transpose_kernel<<<64, 256, 0, stream>>>(root1, root1t, 1);
    rgcn_transpose_kernel<<<512, 256, 0, stream>>>(W2, W2t, NREL);
    rgcn_transpose_kernel<<<64, 256, 0, stream>>>(root2, root2t, 1);
    rgcn_transpose_kernel<<<64, 256, 0, stream>>>(linW, linWt, 1);

    // (dst, rel) edge counts — shared by both layers
    rgcn_zero_kernel<<<512, 256, 0, stream>>>(cnt, NN * NREL);
    rgcn_count_kernel<<<(NE + 255) / 256, 256, 0, stream>>>(dst, et, cnt);

    // layer 1 + ReLU
    run_layer(x, W1t, root1t, b1, src, dst, et, cnt, agg, h1, stream);
    rgcn_relu_kernel<<<(NN * CH + 255) / 256, 256, 0, stream>>>(h1, NN * CH);

    // layer 2
    run_layer(h1, W2t, root2t, b2, src, dst, et, cnt, agg, h2, stream);

    // head: out = h2 @ lin_W + lin_b
    rgcn_gemm_wmma<<<NBLK, dim3(32, 8), 0, stream>>>(
        h2, linWt, linb, nullptr, 0, (float*)d_out);
}